// DeChunkLayer_27857157881953
// MI455X (gfx1250) — compile-verified
//
#include <hip/hip_runtime.h>
#include <hip/hip_bf16.h>
#include <stdint.h>

typedef __attribute__((ext_vector_type(2))) float v2f;
typedef __attribute__((ext_vector_type(8))) float v8f;

#define B_  4
#define L_  4096
#define M_  1024
#define D_  2048
#define EPS_ 1e-4f

// ---------------------------------------------------------------------------
// Kernel 1: per-batch boundary prefix scan, chunk_idx and p_chunked scatter
// ---------------------------------------------------------------------------
__global__ __launch_bounds__(1024)
void dechunk_prep(const float* __restrict__ bprob,      // (B,L,2)
                  const uint8_t* __restrict__ bmask,    // (B,L) bool
                  int* __restrict__ chunk_idx,          // (B,L)
                  float* __restrict__ p_chunked)        // (B,M)
{
  const int b = blockIdx.x;
  const int t = threadIdx.x;            // 1024 threads, 4 tokens each
  __shared__ int s[1024];
  __shared__ int totalS;

  const uint8_t* mrow = bmask + (size_t)b * L_;
  const float*   prow = bprob + (size_t)b * L_ * 2;

  int m[4];
  int lsum = 0;
#pragma unroll
  for (int j = 0; j < 4; ++j) { m[j] = mrow[t * 4 + j] ? 1 : 0; lsum += m[j]; }

  // Hillis-Steele inclusive scan over per-thread sums
  s[t] = lsum;
  __syncthreads();
  for (int off = 1; off < 1024; off <<= 1) {
    int v = (t >= off) ? s[t - off] : 0;
    __syncthreads();
    s[t] += v;
    __syncthreads();
  }
  int incl = s[t];
  if (t == 1023) totalS = incl;
  __syncthreads();
  const int total = totalS;             // # boundary tokens in this batch row
  int run = incl - lsum;                // exclusive prefix for this thread

#pragma unroll
  for (int j = 0; j < 4; ++j) {
    const int l = t * 4 + j;
    run += m[j];                        // inclusive cumsum at token l
    int ci = run - 1;
    ci = ci < 0 ? 0 : (ci > (M_ - 1) ? (M_ - 1) : ci);
    chunk_idx[(size_t)b * L_ + l] = ci;

    float p = prow[l * 2 + 1];
    p = fminf(fmaxf(p, EPS_), 1.0f - EPS_);
    if (m[j]) {                         // boundary: stable rank = run-1
      const int pos = run - 1;
      if (pos < M_) p_chunked[(size_t)b * M_ + pos] = p;
    } else {                            // non-boundary: after all boundaries
      const int pos = total + (l - run);
      if (pos < M_) p_chunked[(size_t)b * M_ + pos] = p;
    }
  }
}

// ---------------------------------------------------------------------------
// Kernel 2: chunked linear recurrence via V_WMMA_F32_16X16X4_F32
//   out[m] = aa[m]*out[m-1] + bb[m],  aa = 1-p (aa[0]=0),  bb = p*x (bb[0]=x[0])
//   Per 16-chunk: OUT = (W.diag(pk)) x X + cumprod (x) carry
//   Branchless suffix-product build of W (no divergent inner loops).
//   One wave per (batch, 16-wide D slice); 64 serial chunks.
// ---------------------------------------------------------------------------
__global__ __launch_bounds__(32)
void dechunk_scan_wmma(const float* __restrict__ x,          // (B,M,D)
                       const float* __restrict__ p_chunked,  // (B,M)
                       float* __restrict__ expanded)         // (B,M,D)
{
  const int slices = D_ / 16;
  const int b     = blockIdx.x / slices;
  const int d0    = (blockIdx.x % slices) * 16;
  const int lane  = threadIdx.x;
  const int half  = lane >> 4;          // 0: lanes 0-15, 1: lanes 16-31
  const int n     = lane & 15;          // col for B/C/D, row M for A

  __shared__ float aaS[16];
  __shared__ float pkS[16];

  float carry = 0.0f;                   // out[m-1] for this lane's column n

  for (int c = 0; c < M_ / 16; ++c) {
    const int m0 = c * 16;
    if (lane < 16) {
      float p  = p_chunked[(size_t)b * M_ + m0 + lane];
      float pk = p, aa = 1.0f - p;
      if (c == 0 && lane == 0) { pk = 1.0f; aa = 0.0f; }  // EMA init row
      aaS[lane] = aa;
      pkS[lane] = pk;
    }
    __syncthreads();

    // cumulative products g[t] = prod_{j<=t} aa[j]  (for the carry term)
    float cp[16];
    cp[0] = aaS[0];
#pragma unroll
    for (int j = 1; j < 16; ++j) cp[j] = cp[j - 1] * aaS[j];

    // Branchless suffix-product row build:
    //   w[k] = pk[k] * prod_{j=k+1..m} aa[j]  (k<=m), else 0;  m = lane&15
    float w[16];
    {
      float r = 0.0f;
#pragma unroll
      for (int k = 15; k >= 0; --k) {
        if (k == n) r = 1.0f;           // v_cndmask, not a branch
        w[k] = r * pkS[k];
        r *= aaS[k];
      }
    }

    // A fragments: 16x4 f32 layout -> M = lane&15, K = 4*kb + 2*half + v
    v2f afrag[4];
#pragma unroll
    for (int kb = 0; kb < 4; ++kb) {
#pragma unroll
      for (int v = 0; v < 2; ++v) {
        afrag[kb][v] = half ? w[kb * 4 + 2 + v] : w[kb * 4 + v];
      }
    }

    // B fragments: pure loads, X[k, nn] = x[b, m0+k, d0+nn]
    v2f bfrag[4];
#pragma unroll
    for (int kb = 0; kb < 4; ++kb) {
#pragma unroll
      for (int v = 0; v < 2; ++v) {
        const int k = kb * 4 + 2 * half + v;
        bfrag[kb][v] = x[((size_t)b * M_ + m0 + k) * D_ + d0 + n];
      }
    }

    // C init = g[m] * carry   (C/D layout: VGPR r -> row r + 8*half, col n)
    v8f cfrag;
#pragma unroll
    for (int r = 0; r < 8; ++r) {
      const float g = half ? cp[r + 8] : cp[r];
      cfrag[r] = g * carry;
    }

    // 4 chained K=4 WMMAs -> K=16
#pragma unroll
    for (int kb = 0; kb < 4; ++kb) {
      cfrag = __builtin_amdgcn_wmma_f32_16x16x4_f32(
          false, afrag[kb], false, bfrag[kb],
          (short)0, cfrag, false, false);
    }

    // store chunk rows
#pragma unroll
    for (int r = 0; r < 8; ++r) {
      const int mm = r + 8 * half;
      expanded[((size_t)b * M_ + m0 + mm) * D_ + d0 + n] = cfrag[r];
    }

    // carry for next chunk = row m=15 value (lives in cfrag[7], lanes 16-31)
    carry = __shfl(cfrag[7], 16 + n, 32);
    __syncthreads();
  }
}

// ---------------------------------------------------------------------------
// Kernel 3: token gather via async LDS DMA (gfx1250 path):
//   out[b,l,:] = expanded[b, chunk_idx[b,l], :]
//   Each wave copies one 8KB row: 16x global_load_async_to_lds_b128,
//   s_wait_asynccnt, 16x global_store_async_from_lds_b128.
// ---------------------------------------------------------------------------
__global__ __launch_bounds__(256)
void dechunk_gather_async(const float* __restrict__ expanded, // (B,M,D)
                          const int* __restrict__ chunk_idx,  // (B,L)
                          float* __restrict__ out)            // (B,L,D)
{
  __shared__ __align__(16) unsigned char smem[8 * 8192];  // 8 waves x 8KB row

  const int lane = threadIdx.x & 31;
  const int wave = threadIdx.x >> 5;
  const int row  = blockIdx.x * 8 + wave;      // 0 .. B*L-1
  const int b = row / L_;
  const int l = row % L_;
  const int ci = chunk_idx[(size_t)b * L_ + l];

  const char* src = (const char*)(expanded + ((size_t)b * M_ + ci) * D_);
  char*       dst = (char*)(out + ((size_t)b * L_ + l) * D_);

  // flat LDS address: low 32 bits of the generic pointer are the LDS offset
  const unsigned lds_base =
      (unsigned)(size_t)(&smem[0]) + (unsigned)(wave * 8192 + lane * 16);

  // memory -> LDS, 512B per issue (32 lanes x 16B)
#pragma unroll
  for (int i = 0; i < 16; ++i) {
    const unsigned lds_off = lds_base + i * 512;
    const unsigned long long ga =
        (unsigned long long)(src + i * 512 + lane * 16);
    asm volatile("global_load_async_to_lds_b128 %0, %1, off"
                 :: "v"(lds_off), "v"(ga) : "memory");
  }
  asm volatile("s_wait_asynccnt 0" ::: "memory");

  // LDS -> memory
#pragma unroll
  for (int i = 0; i < 16; ++i) {
    const unsigned lds_off = lds_base + i * 512;
    const unsigned long long ga =
        (unsigned long long)(dst + i * 512 + lane * 16);
    asm volatile("global_store_async_from_lds_b128 %0, %1, off"
                 :: "v"(ga), "v"(lds_off) : "memory");
  }
  asm volatile("s_wait_asynccnt 0" ::: "memory");
}

// ---------------------------------------------------------------------------
extern "C" void kernel_launch(void* const* d_in, const int* in_sizes, int n_in,
                              void* d_out, int out_size, void* d_ws, size_t ws_size,
                              hipStream_t stream) {
  const float*   x     = (const float*)d_in[0];     // chunked_states (B,M,D) f32
  const float*   bprob = (const float*)d_in[1];     // boundary_prob (B,L,2) f32
  const uint8_t* bmask = (const uint8_t*)d_in[2];   // boundary_mask (B,L) bool
  float* out = (float*)d_out;                       // (B,L,D) f32

  char* ws = (char*)d_ws;
  float* expanded  = (float*)ws;                                        // B*M*D f32
  int*   chunk_idx = (int*)(ws + (size_t)B_ * M_ * D_ * sizeof(float)); // B*L i32
  float* p_chunked = (float*)((char*)chunk_idx + (size_t)B_ * L_ * sizeof(int)); // B*M

  dechunk_prep<<<B_, 1024, 0, stream>>>(bprob, bmask, chunk_idx, p_chunked);
  dechunk_scan_wmma<<<B_ * (D_ / 16), 32, 0, stream>>>(x, p_chunked, expanded);
  dechunk_gather_async<<<(B_ * L_) / 8, 256, 0, stream>>>(expanded, chunk_idx, out);
}